// ThesisMarkowitz_63436666962539
// MI455X (gfx1250) — compile-verified
//
#include <hip/hip_runtime.h>
#include <math.h>

// MI455X / gfx1250: wave32, WMMA 16x16x4 f32, 320KB LDS per WGP.
// One workgroup (128 thr = 4 waves) per batch item. A = gamma*covmat staged
// into LDS once (HBM touched exactly once: 134MB total at 23.3 TB/s ~ 6us);
// all 300 PGA iterations run out of LDS/registers.
//
// Layout: matvec Aw = A*w via v_wmma_f32_16x16x4_f32 with w replicated in all
// 16 B columns (each wave: 2 M-tiles x 32 K-tiles). Elementwise + bisection
// phase is wave-redundant: each lane owns 4 assets in registers, so the
// 40-step bisection (the serial critical path) needs zero barriers.
// Only 2 workgroup barriers per PGA iteration.

typedef float v2f __attribute__((ext_vector_type(2)));
typedef float v8f __attribute__((ext_vector_type(8)));

#define N_ASSETS 128
#define BATCH    2048
#define N_ITERS  300
#define N_BISECT 40
#define LR       0.02f
#define EPSI     1e-8f
#define CAP      1.0f
#define A_STRIDE 132   // 128 + 4 pad: conflict-free ds_load_b64 A-tile fetches

__device__ __forceinline__ float clipc(float x) {
    return fminf(fmaxf(x, -CAP), CAP);
}

__device__ __forceinline__ float wave_sum(float x) {
    #pragma unroll
    for (int off = 16; off > 0; off >>= 1) x += __shfl_xor(x, off, 32);
    return x;
}
__device__ __forceinline__ float wave_min(float x) {
    #pragma unroll
    for (int off = 16; off > 0; off >>= 1) x = fminf(x, __shfl_xor(x, off, 32));
    return x;
}
__device__ __forceinline__ float wave_max(float x) {
    #pragma unroll
    for (int off = 16; off > 0; off >>= 1) x = fmaxf(x, __shfl_xor(x, off, 32));
    return x;
}

// Extract D[r] from the 8-VGPR f32 accumulator with a cndmask chain.
__device__ __forceinline__ float sel8(v8f d, int r) {
    float x = d[0];
    x = (r == 1) ? d[1] : x;
    x = (r == 2) ? d[2] : x;
    x = (r == 3) ? d[3] : x;
    x = (r == 4) ? d[4] : x;
    x = (r == 5) ? d[5] : x;
    x = (r == 6) ? d[6] : x;
    x = (r == 7) ? d[7] : x;
    return x;
}

__global__ __launch_bounds__(128)
void markowitz_pga_wmma(const float* __restrict__ rets,
                        const float* __restrict__ covmat,
                        const float* __restrict__ gammav,
                        const float* __restrict__ alphav,
                        float* __restrict__ out)
{
    extern __shared__ float smem[];
    float* sA  = smem;                      // 128 rows * A_STRIDE
    float* sW  = smem + 128 * A_STRIDE;     // 128  (current weights, WMMA B src)
    float* sAw = sW + 128;                  // 128  (A*w)

    const int t    = threadIdx.x;           // 0..127
    const int lane = t & 31;
    const int wave = t >> 5;                // 0..3
    const int half = lane >> 4;             // 0/1 (lane group for WMMA K-split)
    const int lm   = lane & 15;
    const int b    = blockIdx.x;

    const float gsc   = gammav[b];
    const float a_abs = fabsf(alphav[b]);

    // Per-lane 4-asset slice (assets 4*lane .. 4*lane+3), redundant per wave.
    const float4 ret4 = *(const float4*)(rets + (size_t)b * N_ASSETS + 4 * lane);

    // ---- Stage A = gamma * covmat into LDS (coalesced b128 loads) ----
    const float4* cm4 = (const float4*)(covmat + (size_t)b * N_ASSETS * N_ASSETS);
    #pragma unroll 4
    for (int k = 0; k < 32; ++k) {
        int idx4 = k * 128 + t;             // float4 index 0..4095
        float4 v = cm4[idx4];
        v.x *= gsc; v.y *= gsc; v.z *= gsc; v.w *= gsc;
        int i = idx4 >> 5;                  // row
        int j = (idx4 & 31) << 2;           // col (multiple of 4 -> 16B aligned)
        *(float4*)&sA[i * A_STRIDE + j] = v;
    }

    const float w0 = 1.0f / (float)N_ASSETS;
    float4 w4 = make_float4(w0, w0, w0, w0);
    if (t < N_ASSETS) sW[t] = w0;
    __syncthreads();

    // Wave handles rows [wave*32, wave*32+31] as two 16-row M-tiles.
    const int m0 = wave * 32 + lm;
    const float* A0 = &sA[m0 * A_STRIDE + half * 2];
    const float* A1 = A0 + 16 * A_STRIDE;

    #pragma unroll 1
    for (int it = 0; it < N_ITERS; ++it) {
        // ---- Aw = A*w via v_wmma_f32_16x16x4_f32 (w replicated in B cols) ----
        v8f acc0 = {};
        v8f acc1 = {};
        #pragma unroll 8
        for (int kt = 0; kt < 32; ++kt) {
            const int kb = kt * 4;
            // B(4x16): lanes 0-15 hold K=kb+0/kb+1, lanes 16-31 K=kb+2/kb+3
            v2f bv = *(const v2f*)&sW[kb + half * 2];   // same-address broadcast
            v2f a0 = *(const v2f*)&A0[kb];              // ds_load_2addr_b64 pairs
            v2f a1 = *(const v2f*)&A1[kb];
            acc0 = __builtin_amdgcn_wmma_f32_16x16x4_f32(
                false, a0, false, bv, (short)0, acc0, false, false);
            acc1 = __builtin_amdgcn_wmma_f32_16x16x4_f32(
                false, a1, false, bv, (short)0, acc1, false, false);
        }
        // D columns are identical; lanes {0..7, 16..23} scatter one row each.
        if (lm < 8) {
            int row = wave * 32 + half * 8 + lm;
            sAw[row]      = sel8(acc0, lm);
            sAw[row + 16] = sel8(acc1, lm);
        }
        __syncthreads();   // B1: Aw visible to all waves

        // ---- wave-redundant elementwise phase (no more barriers needed) ----
        float4 Aw4 = *(const float4*)&sAw[4 * lane];    // ds_load_b128

        float p = wave_sum(w4.x * Aw4.x + w4.y * Aw4.y +
                           w4.z * Aw4.z + w4.w * Aw4.w);
        float q = wave_sum(w4.x * w4.x + w4.y * w4.y +
                           w4.z * w4.z + w4.w * w4.w);
        float risk = sqrtf(p + EPSI);
        float nrm  = sqrtf(q + EPSI);

        float4 v4;
        v4.x = w4.x + LR * (ret4.x - Aw4.x / risk - a_abs * (w4.x / nrm));
        v4.y = w4.y + LR * (ret4.y - Aw4.y / risk - a_abs * (w4.y / nrm));
        v4.z = w4.z + LR * (ret4.z - Aw4.z / risk - a_abs * (w4.z / nrm));
        v4.w = w4.w + LR * (ret4.w - Aw4.w / risk - a_abs * (w4.w / nrm));

        // ---- capped-simplex projection: in-register bisection ----
        float vmin = wave_min(fminf(fminf(v4.x, v4.y), fminf(v4.z, v4.w)));
        float vmax = wave_max(fmaxf(fmaxf(v4.x, v4.y), fmaxf(v4.z, v4.w)));
        float lo = vmin - CAP - 1.0f;
        float hi = vmax + CAP;
        #pragma unroll 1
        for (int bi = 0; bi < N_BISECT; ++bi) {
            float mid = 0.5f * (lo + hi);
            float s = clipc(v4.x - mid) + clipc(v4.y - mid)
                    + clipc(v4.z - mid) + clipc(v4.w - mid);
            s = wave_sum(s);
            bool big = s > 1.0f;
            lo = big ? mid : lo;
            hi = big ? hi : mid;
        }
        float tau = 0.5f * (lo + hi);
        w4.x = clipc(v4.x - tau);
        w4.y = clipc(v4.y - tau);
        w4.z = clipc(v4.z - tau);
        w4.w = clipc(v4.w - tau);

        // Only wave 0 publishes w to LDS (WMMA B-operand source for next iter).
        if (wave == 0) *(float4*)&sW[4 * lane] = w4;
        __syncthreads();   // B2: new w visible before next matvec
    }

    if (wave == 0) {
        *(float4*)(out + (size_t)b * N_ASSETS + 4 * lane) = w4;
    }
}

extern "C" void kernel_launch(void* const* d_in, const int* in_sizes, int n_in,
                              void* d_out, int out_size, void* d_ws, size_t ws_size,
                              hipStream_t stream) {
    (void)in_sizes; (void)n_in; (void)out_size; (void)d_ws; (void)ws_size;
    const float* rets   = (const float*)d_in[0];
    const float* covmat = (const float*)d_in[1];
    const float* gammav = (const float*)d_in[2];
    const float* alphav = (const float*)d_in[3];
    float* out = (float*)d_out;

    const size_t shmem = (size_t)(128 * A_STRIDE + 2 * 128) * sizeof(float); // ~68 KB
    (void)hipFuncSetAttribute((const void*)markowitz_pga_wmma,
                              hipFuncAttributeMaxDynamicSharedMemorySize,
                              (int)shmem);
    markowitz_pga_wmma<<<dim3(BATCH), dim3(128), shmem, stream>>>(
        rets, covmat, gammav, alphav, out);
}